// DCE_48249662603417
// MI455X (gfx1250) — compile-verified
//
#include <hip/hip_runtime.h>
#include <math.h>

#define NPIX   102400      // 320*320
#define LCONV  589         // round(58.8/0.1)+1
#define KPAD   592         // LCONV padded to multiple of 4 (WMMA K-step)
#define NSAMP  50
#define DELAYN 30          // int(3/0.1)
#define REVSZ  1184        // reversed AIF image: rev[q]=aif[588-q], zero tail

typedef float v2f __attribute__((ext_vector_type(2)));
typedef float v8f __attribute__((ext_vector_type(8)));

__device__ __forceinline__ float nan0f(float x) { return (x != x) ? 0.0f : x; }

// ---------------------------------------------------------------------------
// Prep kernel (1 block): replicate jnp.interp AIF resampling, the 3s delay,
// and searchsorted(t_samp, sample_time). Writes aif[KPAD] + idx[64] to ws.
// ---------------------------------------------------------------------------
__global__ __launch_bounds__(256) void dce_prep(const float* __restrict__ st,
                                                const float* __restrict__ cp,
                                                float* __restrict__ aif_g,
                                                int*   __restrict__ idx_g)
{
    __shared__ float st_s[NSAMP], cp_s[NSAMP];
    int tid = threadIdx.x;
    if (tid < NSAMP) { st_s[tid] = st[tid]; cp_s[tid] = cp[tid]; }
    __syncthreads();

    for (int t = tid; t < KPAD; t += 256) {
        float v = 0.0f;
        if (t >= DELAYN && t < LCONV) {
            float x = (float)(t - DELAYN) * 0.1f;       // t_samp[t-DELAY], same fp math as jnp
            if (x <= st_s[0])              v = cp_s[0];
            else if (x >= st_s[NSAMP - 1]) v = cp_s[NSAMP - 1];
            else {
                int i = 0;
                for (int j = 1; j < NSAMP - 1; ++j)
                    if (st_s[j] <= x) i = j;            // last interval with st[i] <= x
                v = cp_s[i] + (cp_s[i + 1] - cp_s[i]) * (x - st_s[i]) / (st_s[i + 1] - st_s[i]);
            }
        }
        aif_g[t] = v;
    }

    if (tid < 64) {
        int id = -2;                                    // padded rows -> all-zero A rows
        if (tid < NSAMP) {
            float xv = st_s[tid];
            int j = LCONV - 1;                          // JAX gather clamps OOB -> 588
            for (int t = LCONV - 1; t >= 0; --t)
                if ((float)t * 0.1f >= xv) j = t;       // searchsorted 'left'
            id = j;
        }
        idx_g[tid] = id;
    }
}

// ---------------------------------------------------------------------------
// Main kernel: sampled convolution as 64x592 * 592x102400 f32 GEMM on WMMA.
// One wave per 16-pixel strip, 4 M-tiles (64 sample rows) per wave.
// B (per-pixel kernel h[t] = P*rm^t + Q*rp^t) regenerated in registers via
// geometric recurrences; A = banded triangular matrix of AIF samples read as
// ascending pairs from a REVERSED zero-tailed LDS image so ds_load_2addr_b32
// lands the pair in WMMA register order (no swap movs, no exec masking).
// ---------------------------------------------------------------------------
__device__ __forceinline__ void write_tile(v8f acc, int m, int hl, int p,
                                           float M0t, float COSF, float offs,
                                           float* __restrict__ out)
{
    union { v8f v; float f[8]; } u; u.v = acc;
#pragma unroll
    for (int jj = 0; jj < 8; ++jj) {
        int s = m * 16 + jj + 8 * hl;                   // C layout: VGPR jj -> M=jj / jj+8
        if (s < NSAMP) {
            float CA   = u.f[jj];
            float E1CA = __expf(-0.00487f * (1.0f + 4.3f * CA));   // exp(-TR*(R1+R1CA*CA))
            float CAt  = M0t * (1.0f - E1CA) *
                         __builtin_amdgcn_rcpf(1.0f - E1CA * COSF); // v_rcp_f32, no div expansion
            out[s * NPIX + p] = CAt + offs;
        }
    }
}

__global__ __launch_bounds__(256) void dce_main(const float* __restrict__ param,
                                                const float* __restrict__ aif_g,
                                                const int*   __restrict__ idx_g,
                                                float* __restrict__ out)
{
    __shared__ float rev_s[REVSZ];      // rev[q] = aif[588-q] for q<=588, else 0
    __shared__ int   idx_s[64];
    int tid = threadIdx.x;
    for (int i = tid; i < REVSZ; i += 256) {
        int t = (LCONV - 1) - i;                        // 588 - q
        rev_s[i] = (t >= 0) ? aif_g[t] : 0.0f;
    }
    if (tid < 64) idx_s[tid] = idx_g[tid];
    __syncthreads();

    const int wave = tid >> 5;
    const int lane = tid & 31;
    const int hl   = lane >> 4;          // half-wave: K split {0,1} vs {2,3}
    const int lpix = lane & 15;
    const int tile = blockIdx.x * 8 + wave;
    const int p    = tile * 16 + lpix;

    // ---- per-pixel 2CXM coefficients (both half-waves compute identically) ----
    float ve = param[0 * NPIX + p];
    float vp = param[1 * NPIX + p];
    float fp = param[2 * NPIX + p];
    float ps = param[3 * NPIX + p];

    float Te = nan0f(ve / ps);
    float T  = nan0f((vp + ve) / fp);
    float Tc = nan0f(vp / fp);
    float sm   = T + Te;
    float disc = sqrtf(sm * sm - 4.0f * Tc * Te);
    float inv2 = 1.0f / (2.0f * Tc * Te);
    float thp  = nan0f((sm + disc) * inv2);
    float thm  = nan0f((sm - disc) * inv2);
    float pref = thp * thm;
    float dn   = thp - thm;
    float am = pref / dn;                       // he:  am*rm^t + ap*rp^t
    float ap = -am;
    float bm = pref * (1.0f - Te * thm) / dn;   // hp:  bm*rm^t + bp*rp^t
    float bp = pref * (Te * thp - 1.0f) / dn;

    float rm    = __expf(-0.1f  * thm);
    float rp    = __expf(-0.1f  * thp);
    float rm589 = __expf(-58.9f * thm);         // rm^589 (closed-form geometric sum)
    float rp589 = __expf(-58.9f * thp);
    float Smm = (1.0f - rm589) / (1.0f - rm);
    float Spp = (1.0f - rp589) / (1.0f - rp);
    float sumHe = am * Smm + ap * Spp;
    float sumHp = bm * Smm + bp * Spp;

    // combined normalized kernel: h[t] = P*rm^t + Q*rp^t  (conv is linear)
    float P = nan0f(vp * nan0f(bm / sumHp) + ve * nan0f(am / sumHe));
    float Q = nan0f(vp * nan0f(bp / sumHp) + ve * nan0f(ap / sumHe));

    // running powers at this lane's K phase: t0 = 4k + 2*hl
    float rm2 = rm * rm, rp2 = rp * rp;
    float rm4 = rm2 * rm2, rp4 = rp2 * rp2;
    float rmt = hl ? rm2 : 1.0f;
    float rpt = hl ? rp2 : 1.0f;

    // reversed-image indices: q = 588 - (idx_s - t0); ascending by 4 per k-step.
    // Accesses q and q+1 stay inside [0, REVSZ) by construction (idx >= -2).
    int q0 = (LCONV - 1) - idx_s[ 0 + lpix] + 2 * hl;
    int q1 = (LCONV - 1) - idx_s[16 + lpix] + 2 * hl;
    int q2 = (LCONV - 1) - idx_s[32 + lpix] + 2 * hl;
    int q3 = (LCONV - 1) - idx_s[48 + lpix] + 2 * hl;

    v8f acc0 = {}, acc1 = {}, acc2 = {}, acc3 = {};

#pragma unroll 4
    for (int k = 0; k < KPAD / 4; ++k) {
        // B fragment: h at t0 and t0+1 for this lane's pixel
        float b0 = P * rmt + Q * rpt;
        float rmt1 = rmt * rm, rpt1 = rpt * rp;
        float b1 = P * rmt1 + Q * rpt1;
        rmt *= rm4; rpt *= rp4;
        v2f B; B.x = b0; B.y = b1;

        // A fragments: ascending pair loads, already in WMMA register order
        v2f A0, A1, A2, A3;
        A0.x = rev_s[q0];  A0.y = rev_s[q0 + 1];
        A1.x = rev_s[q1];  A1.y = rev_s[q1 + 1];
        A2.x = rev_s[q2];  A2.y = rev_s[q2 + 1];
        A3.x = rev_s[q3];  A3.y = rev_s[q3 + 1];
        q0 += 4; q1 += 4; q2 += 4; q3 += 4;

        acc0 = __builtin_amdgcn_wmma_f32_16x16x4_f32(false, A0, false, B, (short)0, acc0, false, false);
        acc1 = __builtin_amdgcn_wmma_f32_16x16x4_f32(false, A1, false, B, (short)0, acc1, false, false);
        acc2 = __builtin_amdgcn_wmma_f32_16x16x4_f32(false, A2, false, B, (short)0, acc2, false, false);
        acc3 = __builtin_amdgcn_wmma_f32_16x16x4_f32(false, A3, false, B, (short)0, acc3, false, false);
    }

    // ---- SPGR signal transform (uniform scalars) ----
    const float COSF = 0.98480775301220805936f;   // cos(10deg)
    const float SINF = 0.17364817766693034885f;   // sin(10deg)
    float E1   = __expf(-0.00487f);               // exp(-TR*R1)
    float M0   = 100.0f * (1.0f - COSF * E1) / (SINF * (1.0f - E1));
    float M0t  = M0 * SINF;
    float Mst  = M0t * (1.0f - E1) / (1.0f - E1 * COSF);
    float offs = 100.0f - Mst;                    // SIG_BASELINE - M_steady

    write_tile(acc0, 0, hl, p, M0t, COSF, offs, out);
    write_tile(acc1, 1, hl, p, M0t, COSF, offs, out);
    write_tile(acc2, 2, hl, p, M0t, COSF, offs, out);
    write_tile(acc3, 3, hl, p, M0t, COSF, offs, out);
}

// ---------------------------------------------------------------------------
extern "C" void kernel_launch(void* const* d_in, const int* in_sizes, int n_in,
                              void* d_out, int out_size, void* d_ws, size_t ws_size,
                              hipStream_t stream)
{
    (void)in_sizes; (void)n_in; (void)out_size; (void)ws_size;
    const float* param = (const float*)d_in[0];
    const float* st    = (const float*)d_in[1];
    const float* cp    = (const float*)d_in[2];
    float* out   = (float*)d_out;
    float* aif_g = (float*)d_ws;                                   // KPAD floats
    int*   idx_g = (int*)((char*)d_ws + KPAD * sizeof(float));     // 64 ints

    dce_prep<<<1, 256, 0, stream>>>(st, cp, aif_g, idx_g);
    // 102400 pixels / 16 per wave = 6400 waves; 8 waves per block -> 800 blocks
    dce_main<<<800, 256, 0, stream>>>(param, aif_g, idx_g, out);
}